// SketchCritic_83382495084727
// MI455X (gfx1250) — compile-verified
//
#include <hip/hip_runtime.h>
#include <hip/hip_bf16.h>

// Problem constants (reference: B=128, S=4096, M=20)
#define BB 128
#define SS 4096
#define MM 20
#define ROWS (BB * SS)          // 524288, divisible by 256
#define BLK 256
#define GRID (ROWS / BLK)       // 2048

#define LOG_2PI 1.8378770664093453f

typedef __attribute__((ext_vector_type(2))) float v2f;
typedef __attribute__((ext_vector_type(8))) float v8f;

// Wave32 full-lane sum using the matrix pipe:
// A (16x4 f32): lanes 0-15 -> {VGPR0:K0, VGPR1:K1}, lanes 16-31 -> {K2,K3}.
// Put lane value in VGPR0, zero in VGPR1, B = ones(4x16):
//   D[m][n] = val[m] + val[m+16]  (replicated across n)
// Per-lane sum of the 8 D VGPRs gives half-sums; shfl_xor(16) completes it.
// REQUIRES EXEC == all ones (call with no divergence).
__device__ __forceinline__ float wave_sum_wmma(float v) {
    v2f a; a[0] = v;     a[1] = 0.0f;
    v2f b; b[0] = 1.0f;  b[1] = 1.0f;
    v8f c = {};
    c = __builtin_amdgcn_wmma_f32_16x16x4_f32(
            false, a, false, b, (short)0, c, false, false);
    float s = c[0] + c[1] + c[2] + c[3] + c[4] + c[5] + c[6] + c[7];
    s += __shfl_xor(s, 16, 32);
    return s;   // every lane holds the full 32-lane sum
}

__device__ __forceinline__ float block_sum(float v, float* red) {
    float w = wave_sum_wmma(v);           // all threads active here
    const int wid = threadIdx.x >> 5;
    const int lid = threadIdx.x & 31;
    if (lid == 0) red[wid] = w;
    __syncthreads();
    float total = 0.0f;
    if (threadIdx.x == 0) {
        #pragma unroll
        for (int i = 0; i < BLK / 32; ++i) total += red[i];
    }
    return total;                          // valid on thread 0 only
}

__global__ void __launch_bounds__(BLK)
row_loss_kernel(const float* __restrict__ xs,
                const float* __restrict__ logits,
                const float* __restrict__ mus,
                const float* __restrict__ sigmas,
                const float* __restrict__ pen_pred,
                float* __restrict__ partial) {
    const int r = blockIdx.x * BLK + threadIdx.x;   // grid covers ROWS exactly
    const int s = r & (SS - 1);

    // ---- relative position (prev within same sequence; zero at s==0) ----
    const float* x = xs + (size_t)r * 5;
    const float px = x[0], py = x[1];
    float prevx = 0.0f, prevy = 0.0f;
    if (s != 0) { prevx = x[-5]; prevy = x[-4]; }
    const float rx = px - prevx;
    const float ry = py - prevy;

    const float* lg = logits + (size_t)r * MM;       // 80B rows, 16B aligned
    const float* mu = mus    + (size_t)r * (MM * 2); // 160B rows
    const float* sg = sigmas + (size_t)r * (MM * 3); // 240B rows

    // ---- pass 1: t_m = logit_m + comp_logp_m ; track maxes ----
    // comp = -0.5*(z0^2+z1^2) - LOG_2PI - log(l00) - log(l11)
    //      = -0.5*(z0^2+z1^2) - LOG_2PI + log(r00*r11),  r = v_rcp(l)
    float t[MM], l[MM];
    float m1 = -3.4e38f, m2 = -3.4e38f;
    #pragma unroll
    for (int m = 0; m < MM; ++m) {
        const float l00 = sg[3 * m + 0];
        const float l10 = sg[3 * m + 1];
        const float l11 = sg[3 * m + 2];
        const float r00 = __builtin_amdgcn_rcpf(l00);   // 1 TRANS op
        const float r11 = __builtin_amdgcn_rcpf(l11);   // 1 TRANS op
        const float d0  = rx - mu[2 * m + 0];
        const float d1  = ry - mu[2 * m + 1];
        const float z0  = d0 * r00;
        const float z1  = (d1 - l10 * z0) * r11;
        const float comp = fmaf(-0.5f, fmaf(z0, z0, z1 * z1),
                                __logf(r00 * r11) - LOG_2PI);
        const float li = lg[m];
        t[m] = li + comp;
        l[m] = li;
        m1 = fmaxf(m1, t[m]);
        m2 = fmaxf(m2, li);
    }
    // ---- pass 2: double logsumexp.
    // LSE(log_softmax(lg)+comp) == LSE(lg+comp) - LSE(lg)
    float e1 = 0.0f, e2 = 0.0f;
    #pragma unroll
    for (int m = 0; m < MM; ++m) {
        e1 += __expf(t[m] - m1);
        e2 += __expf(l[m] - m2);
    }
    const float mix_logp = (m1 + __logf(e1)) - (m2 + __logf(e2));

    // ---- pen loss term: -sum_j pen_pred[j] * log_softmax(pen_true)[j] ----
    const float p0 = x[2], p1 = x[3], p2 = x[4];
    const float pmax = fmaxf(p0, fmaxf(p1, p2));
    const float plse = pmax + __logf(__expf(p0 - pmax) + __expf(p1 - pmax)
                                     + __expf(p2 - pmax));
    const float* pp = pen_pred + (size_t)r * 3;
    const float pen_row = -(pp[0] * (p0 - plse) + pp[1] * (p1 - plse)
                            + pp[2] * (p2 - plse));

    const float v = -mix_logp + pen_row * (1.0f / (float)ROWS);

    // ---- deterministic block reduction (WMMA wave sum + LDS) ----
    __shared__ float red[BLK / 32];
    const float bsum = block_sum(v, red);
    if (threadIdx.x == 0) partial[blockIdx.x] = bsum;
}

__global__ void __launch_bounds__(BLK)
final_reduce_kernel(const float* __restrict__ partial, int n,
                    float* __restrict__ out) {
    float v = 0.0f;
    for (int i = threadIdx.x; i < n; i += BLK) v += partial[i];
    __shared__ float red[BLK / 32];
    const float total = block_sum(v, red);   // all threads re-converged here
    if (threadIdx.x == 0) out[0] = total;
}

extern "C" void kernel_launch(void* const* d_in, const int* in_sizes, int n_in,
                              void* d_out, int out_size, void* d_ws, size_t ws_size,
                              hipStream_t stream) {
    const float* xs       = (const float*)d_in[0];
    const float* logits   = (const float*)d_in[1];
    const float* mus      = (const float*)d_in[2];
    const float* sigmas   = (const float*)d_in[3];
    const float* pen_pred = (const float*)d_in[4];
    float* partial = (float*)d_ws;           // GRID floats = 8 KB scratch
    float* out     = (float*)d_out;

    row_loss_kernel<<<GRID, BLK, 0, stream>>>(xs, logits, mus, sigmas,
                                              pen_pred, partial);
    final_reduce_kernel<<<1, BLK, 0, stream>>>(partial, GRID, out);
}